// EGATConv_16338055594503
// MI455X (gfx1250) — compile-verified
//
#include <hip/hip_runtime.h>
#include <hip/hip_bf16.h>

#define N_NODES  100000
#define N_EDGES  1600000
#define N_GRAPHS 64
#define CIN  128
#define COUT 64
#define EPS  1e-5f

typedef __attribute__((ext_vector_type(2))) float v2f;
typedef __attribute__((ext_vector_type(8))) float v8f;

// ---------------------------------------------------------------------------
// Kernel 1: h = x @ W via V_WMMA_F32_16X16X4_F32.
// One wave computes a 16(M) x 64(N) strip. Weight staged in LDS in a
// K-pair-swizzled layout so every B fragment is one contiguous ds_load_b64:
//   wswz[(k>>1)*128 + n*2 + (k&1)] = w[k][n]
// Fragment for K-pair p, column c: v2f at wswz[p*128 + c*2]  (8B aligned).
// 100000 = 16*6250 exactly -> no M remainder.
// ---------------------------------------------------------------------------
__global__ __launch_bounds__(256) void gemm_wmma_f32(
    const float* __restrict__ x, const float* __restrict__ w,
    float* __restrict__ h, int ntiles) {
  __shared__ float wlds[CIN * COUT];  // 32 KB (WGP has 320 KB)
  const int t = threadIdx.x;
  // stage + swizzle: thread reads w[k][n..n+3], scatters with stride 2
  for (int i = t * 4; i < CIN * COUT; i += 256 * 4) {
    float4 v = *(const float4*)(&w[i]);
    const int k = i >> 6, n = i & 63;
    float* d = &wlds[(k >> 1) * 128 + n * 2 + (k & 1)];
    d[0] = v.x; d[2] = v.y; d[4] = v.z; d[6] = v.w;
  }
  __syncthreads();

  const int wave = t >> 5, lane = t & 31;
  const int tile = blockIdx.x * 8 + wave;
  if (tile >= ntiles) return;                 // wave-uniform: EXEC stays all-1s

  const int mrowA = tile * 16 + (lane & 15);  // A: M = lane%16
  const int koff  = (lane >> 4) << 1;         // K offset 0 / 2 for lane halves
  const int n0    = lane & 15;                // B/D: N = lane%16

  // loop-invariant LDS base: K-pair select (128 floats) + column*2
  const float* bbase = &wlds[((lane >> 4) << 7) + n0 * 2];

  v8f acc0 = {}, acc1 = {}, acc2 = {}, acc3 = {};
  const float* xrow = x + (long)mrowA * CIN + koff;

  for (int kb = 0; kb < CIN; kb += 4) {
    v2f a  = *(const v2f*)(xrow + kb);            // K = kb+koff, kb+koff+1
    const float* bp = bbase + kb * 64;            // kb*64 folds into DS offset
    v2f b0 = *(const v2f*)(bp +  0);
    v2f b1 = *(const v2f*)(bp + 32);
    v2f b2 = *(const v2f*)(bp + 64);
    v2f b3 = *(const v2f*)(bp + 96);
    acc0 = __builtin_amdgcn_wmma_f32_16x16x4_f32(false, a, false, b0, (short)0, acc0, false, false);
    acc1 = __builtin_amdgcn_wmma_f32_16x16x4_f32(false, a, false, b1, (short)0, acc1, false, false);
    acc2 = __builtin_amdgcn_wmma_f32_16x16x4_f32(false, a, false, b2, (short)0, acc2, false, false);
    acc3 = __builtin_amdgcn_wmma_f32_16x16x4_f32(false, a, false, b3, (short)0, acc3, false, false);
  }

  // D layout: lane l, VGPR j -> M = j + 8*(l/16), N = l%16
  const int mrowD = tile * 16 + (lane >> 4) * 8;
#pragma unroll
  for (int j = 0; j < 8; ++j) {
    float* hp = h + (long)(mrowD + j) * COUT + n0;
    hp[0]  = acc0[j];
    hp[16] = acc1[j];
    hp[32] = acc2[j];
    hp[48] = acc3[j];
  }
}

// ---------------------------------------------------------------------------
// Kernel 2: per-graph per-channel mean / rstd (batch_mask is sorted).
// One block per graph; segment bounds via binary search.
// ---------------------------------------------------------------------------
__device__ __forceinline__ int lower_bound_dev(const int* __restrict__ b, int key) {
  int lo = 0, hi = N_NODES;
  while (lo < hi) { int mid = (lo + hi) >> 1; if (b[mid] < key) lo = mid + 1; else hi = mid; }
  return lo;
}

__global__ __launch_bounds__(256) void graph_stats(
    const float* __restrict__ h, const int* __restrict__ batch,
    float* __restrict__ mean, float* __restrict__ rstd) {
  const int g = blockIdx.x;
  const int start = lower_bound_dev(batch, g);
  const int end   = lower_bound_dev(batch, g + 1);
  const int t = threadIdx.x, c = t & 63, sub = t >> 6;

  float s = 0.f, s2 = 0.f;
  for (int n = start + sub; n < end; n += 4) {
    float v = h[(long)n * COUT + c];
    s += v; s2 += v * v;
  }
  __shared__ float sh1[256], sh2[256];
  sh1[t] = s; sh2[t] = s2;
  __syncthreads();
  if (t < 64) {
    float ts  = sh1[t] + sh1[t + 64] + sh1[t + 128] + sh1[t + 192];
    float ts2 = sh2[t] + sh2[t + 64] + sh2[t + 128] + sh2[t + 192];
    int cnt = end - start;
    float m = 0.f, r = 0.f;
    if (cnt > 0) {
      m = ts / (float)cnt;
      float var = fmaxf(ts2 / (float)cnt - m * m, 0.f);
      r = rsqrtf(var + EPS);
    }
    mean[g * 64 + t] = m;
    rstd[g * 64 + t] = r;
  }
}

// ---------------------------------------------------------------------------
// Kernel 3: normalize h in place + per-node attention projections
// s_src[n,k] = hn . alpha_w[k, :64],  s_dst[n,k] = hn . alpha_w[k, 64:]
// ---------------------------------------------------------------------------
__global__ __launch_bounds__(256) void norm_proj(
    float* __restrict__ h, const int* __restrict__ batch,
    const float* __restrict__ mean, const float* __restrict__ rstd,
    const float* __restrict__ aw, float* __restrict__ ssrc, float* __restrict__ sdst) {
  const int n = blockIdx.x * blockDim.x + threadIdx.x;
  if (n >= N_NODES) return;
  const int g = batch[n];
  const float* mg = mean + g * 64;
  const float* rg = rstd + g * 64;
  float* hn = h + (long)n * 64;
  float accs[4] = {0.f, 0.f, 0.f, 0.f};
  float accd[4] = {0.f, 0.f, 0.f, 0.f};
  for (int c = 0; c < 64; c += 4) {
    float4 hv = *(float4*)(hn + c);
    float4 mv = *(const float4*)(mg + c);
    float4 rv = *(const float4*)(rg + c);
    float4 o;
    o.x = (hv.x - mv.x) * rv.x; o.y = (hv.y - mv.y) * rv.y;
    o.z = (hv.z - mv.z) * rv.z; o.w = (hv.w - mv.w) * rv.w;
    *(float4*)(hn + c) = o;
#pragma unroll
    for (int k = 0; k < 4; ++k) {
      const float* wk = aw + k * 128;
      accs[k] += o.x * wk[c] + o.y * wk[c + 1] + o.z * wk[c + 2] + o.w * wk[c + 3];
      accd[k] += o.x * wk[64 + c] + o.y * wk[64 + c + 1] + o.z * wk[64 + c + 2] + o.w * wk[64 + c + 3];
    }
  }
#pragma unroll
  for (int k = 0; k < 4; ++k) { ssrc[n * 4 + k] = accs[k]; sdst[n * 4 + k] = accd[k]; }
}

// ---------------------------------------------------------------------------
// Kernel 4: per-edge alpha (head-mean) + degree accumulation over |alpha|
// ---------------------------------------------------------------------------
__global__ __launch_bounds__(256) void edge_alpha(
    const int* __restrict__ row, const int* __restrict__ col,
    const float* __restrict__ ea, const float* __restrict__ ssrc,
    const float* __restrict__ sdst, const float* __restrict__ ab,
    float* __restrict__ alpha_out, float* __restrict__ deg) {
  const int e = blockIdx.x * blockDim.x + threadIdx.x;
  if (e >= N_EDGES) return;
  const int r = row[e], c = col[e];
  float a;
  if (r == c) {
    a = 1.0f;
  } else {
    float4 sr = *(const float4*)(ssrc + (long)r * 4);
    float4 sd = *(const float4*)(sdst + (long)c * 4);
    float4 bb = *(const float4*)ab;
    float eav = ea[e];
    float s = 1.0f / (1.0f + __expf(-(sr.x + sd.x + bb.x)))
            + 1.0f / (1.0f + __expf(-(sr.y + sd.y + bb.y)))
            + 1.0f / (1.0f + __expf(-(sr.z + sd.z + bb.z)))
            + 1.0f / (1.0f + __expf(-(sr.w + sd.w + bb.w)));
    a = eav * s * 0.25f;
  }
  alpha_out[e] = a;
  atomicAdd(&deg[r], fabsf(a));
}

// ---------------------------------------------------------------------------
// Kernel 5: dinv = deg > 0 ? deg^-1/2 : 0
// ---------------------------------------------------------------------------
__global__ __launch_bounds__(256) void make_dinv(
    const float* __restrict__ deg, float* __restrict__ dinv) {
  const int n = blockIdx.x * blockDim.x + threadIdx.x;
  if (n >= N_NODES) return;
  float d = deg[n];
  dinv[n] = (d > 0.f) ? rsqrtf(d) : 0.f;
}

// ---------------------------------------------------------------------------
// Kernel 6: out[row] += dinv[row]*alpha*dinv[col] * h[col]; 4 threads / edge.
// Prefetch the gathered h[col] cacheline while dinv/alpha loads are in flight.
// ---------------------------------------------------------------------------
__global__ __launch_bounds__(256) void edge_scatter(
    const int* __restrict__ row, const int* __restrict__ col,
    const float* __restrict__ alpha, const float* __restrict__ dinv,
    const float* __restrict__ h, float* __restrict__ out) {
  const long gid = (long)blockIdx.x * blockDim.x + threadIdx.x;
  const long e = gid >> 2;
  if (e >= N_EDGES) return;
  const int sub = (int)(gid & 3);
  const int r = row[e], c = col[e];
  const float* hc = h + (long)c * 64 + sub * 16;
  __builtin_prefetch(hc, 0, 3);               // global_prefetch_b8 (gfx1250)
  const float nrm = dinv[r] * alpha[e] * dinv[c];
  if (nrm == 0.f) return;
  float* op = out + (long)r * 64 + sub * 16;
#pragma unroll
  for (int i = 0; i < 16; i += 4) {
    float4 v = *(const float4*)(hc + i);
    atomicAdd(op + i + 0, nrm * v.x);
    atomicAdd(op + i + 1, nrm * v.y);
    atomicAdd(op + i + 2, nrm * v.z);
    atomicAdd(op + i + 3, nrm * v.w);
  }
}

// ---------------------------------------------------------------------------
// Utility kernels
// ---------------------------------------------------------------------------
__global__ void zero_f32(float* __restrict__ p, long n) {
  long i = (long)blockIdx.x * blockDim.x + threadIdx.x;
  const long stride = (long)gridDim.x * blockDim.x;
  for (; i < n; i += stride) p[i] = 0.f;
}

__global__ void copy_i32(const int* __restrict__ src, int* __restrict__ dst, long n) {
  long i = (long)blockIdx.x * blockDim.x + threadIdx.x;
  const long stride = (long)gridDim.x * blockDim.x;
  for (; i < n; i += stride) dst[i] = src[i];
}

// ---------------------------------------------------------------------------
extern "C" void kernel_launch(void* const* d_in, const int* in_sizes, int n_in,
                              void* d_out, int out_size, void* d_ws, size_t ws_size,
                              hipStream_t stream) {
  const float* x       = (const float*)d_in[0];  // [N,128]
  const float* weight  = (const float*)d_in[1];  // [128,64]
  const float* alpha_w = (const float*)d_in[2];  // [4,128]
  const float* alpha_b = (const float*)d_in[3];  // [4]
  const float* e_attr  = (const float*)d_in[4];  // [E]
  const int*   e_index = (const int*)d_in[5];    // [2,E]
  const int*   batch   = (const int*)d_in[6];    // [N]
  const int* row = e_index;
  const int* col = e_index + N_EDGES;

  // d_out: out [N,64,1] | alpha [E,1] | edge_index [2,E] (int bits)
  float* out_h     = (float*)d_out;
  float* out_alpha = out_h + (long)N_NODES * COUT;
  int*   out_ei    = (int*)(out_alpha + N_EDGES);

  // workspace layout (bytes, 128-aligned)
  char* ws = (char*)d_ws;
  float* h    = (float*)(ws);                               // N*64*4 = 25,600,000
  float* ssrc = (float*)(ws + 25600000);                    // N*4*4  =  1,600,000
  float* sdst = (float*)(ws + 27200000);                    // N*4*4  =  1,600,000
  float* mean = (float*)(ws + 28800000);                    // 64*64*4 =    16,384
  float* rstd = (float*)(ws + 28816384);                    // 64*64*4 =    16,384
  float* deg  = (float*)(ws + 28832768);                    // N*4    =    400,000
  float* dinv = (float*)(ws + 29232768);                    // N*4    =    400,000

  // 0) zero accumulators (d_out is poisoned; determinism across replays)
  zero_f32<<<2048, 256, 0, stream>>>(out_h, (long)N_NODES * COUT);
  zero_f32<<<512, 256, 0, stream>>>(deg, (long)N_NODES);

  // 1) h = x @ W  (WMMA f32)
  const int ntiles = N_NODES / 16;                          // 6250
  gemm_wmma_f32<<<(ntiles + 7) / 8, 256, 0, stream>>>(x, weight, h, ntiles);

  // 2) per-graph stats
  graph_stats<<<N_GRAPHS, 256, 0, stream>>>(h, batch, mean, rstd);

  // 3) normalize + node-level attention projections
  norm_proj<<<(N_NODES + 255) / 256, 256, 0, stream>>>(h, batch, mean, rstd,
                                                       alpha_w, ssrc, sdst);

  // 4) edge alpha + degree
  edge_alpha<<<(N_EDGES + 255) / 256, 256, 0, stream>>>(row, col, e_attr, ssrc,
                                                        sdst, alpha_b, out_alpha, deg);

  // 5) dinv
  make_dinv<<<(N_NODES + 255) / 256, 256, 0, stream>>>(deg, dinv);

  // 6) message scatter
  const long sc_threads = (long)N_EDGES * 4;
  edge_scatter<<<(int)((sc_threads + 255) / 256), 256, 0, stream>>>(
      row, col, out_alpha, dinv, h, out_h);

  // 7) edge_index passthrough (bit-exact int copy)
  copy_i32<<<2048, 256, 0, stream>>>(e_index, out_ei, (long)2 * N_EDGES);
}